// GAT_26164940767885
// MI455X (gfx1250) — compile-verified
//
#include <hip/hip_runtime.h>
#include <hip/hip_bf16.h>
#include <math.h>

#define HEADS 4
#define DIN   256
#define DHID  64
#define DCLS  40

typedef float v2f __attribute__((ext_vector_type(2)));
typedef float v8f __attribute__((ext_vector_type(8)));

// ---------------------------------------------------------------------------
// GEMM: C[M,OUT] = A[M,K] @ B[K,OUT] + bias[OUT]   (fp32, CDNA5 WMMA 16x16x4)
// B is pre-packed pair-major + zero-padded:  Bp[(k>>1)*PW*2 + col*2 + (k&1)]
// so each lane's (K,K+1) B pair is one unguarded global_load_b64.
// Block = 128 threads (4 waves) covering 16 rows; wave w covers cols [64w,64w+64).
// A tile staged in LDS, padded stride 260 (conflict-free ds_load_b64).
// ---------------------------------------------------------------------------
__global__ void __launch_bounds__(128) gat_gemm_wmma(
    const float* __restrict__ A, const float* __restrict__ Bp,
    const float* __restrict__ bias, float* __restrict__ C,
    int M, int K, int PW, int OUT)
{
    const int LDK = 260;
    __shared__ float As[16 * 260];
    const int tid  = threadIdx.x;
    const int wave = tid >> 5;
    const int lane = tid & 31;
    const int m0   = blockIdx.x * 16;

    // cooperative, coalesced float4 load of the 16 x K A tile
    for (int i = 0; i < K / 32; ++i) {
        int f   = (i * 128 + tid) * 4;
        int r   = f / K;
        int k   = f % K;
        int row = m0 + r;
        float4 v = make_float4(0.f, 0.f, 0.f, 0.f);
        if (row < M) v = *(const float4*)(A + (size_t)row * K + k);
        float* p = &As[r * LDK + k];
        p[0] = v.x; p[1] = v.y; p[2] = v.z; p[3] = v.w;
    }
    __syncthreads();

    const int n0 = wave * 64;
    if (n0 >= PW) return;                 // wave-uniform: EXEC stays full below

    const int r    = lane & 15;           // A row / B col within tile
    const int koff = (lane >> 4) * 2;     // lanes 16-31 hold K+2,K+3
    const int cbase = (n0 + r) << 1;      // packed col*2

    v8f acc[4] = {};
#pragma unroll 4
    for (int k0 = 0; k0 < K; k0 += 4) {
        int ka = k0 + koff;               // even
        v2f a = *(const v2f*)(As + r * LDK + ka);
        const float* bpp = Bp + (size_t)ka * PW + cbase;
#pragma unroll
        for (int t = 0; t < 4; ++t) {
            v2f b = *(const v2f*)(bpp + t * 32);      // 16 cols * 2 floats
            acc[t] = __builtin_amdgcn_wmma_f32_16x16x4_f32(
                false, a, false, b, (short)0, acc[t], false, false);
        }
    }

    // C/D layout: VGPR i -> row i (lanes 0-15) / row i+8 (lanes 16-31)
    const int rhi = (lane >> 4) * 8;
#pragma unroll
    for (int t = 0; t < 4; ++t) {
        int col = n0 + t * 16 + (lane & 15);
        if (col >= OUT) continue;
        float bv = bias[col];
#pragma unroll
        for (int i = 0; i < 8; ++i) {
            int row = m0 + rhi + i;
            if (row < M) C[(size_t)row * OUT + col] = acc[t][i] + bv;
        }
    }
}

// pack W[h][k][d] (H x K x hid) -> pair-packed zero-padded panel of width PW:
//   Bp[(k>>1)*PW*2 + col*2 + (k&1)] = (col < H*hid) ? W[h][k][d] : 0
__global__ void gat_pack(const float* __restrict__ W, float* __restrict__ Bp,
                         int K, int hid, int H, int PW)
{
    int t = blockIdx.x * blockDim.x + threadIdx.x;    // output element
    if (t >= K * PW) return;
    int pair = t / (PW * 2);
    int rem  = t % (PW * 2);
    int col  = rem >> 1;
    int k    = pair * 2 + (rem & 1);
    float v  = 0.f;
    if (col < H * hid) {
        int h = col / hid, d = col % hid;
        v = W[((size_t)h * K + k) * hid + d];
    }
    Bp[t] = v;
}

// a1[n,h] = ft[n, h*hid : ] . al[h] + alb[h]   (and a2 with ar/arb)
__global__ void gat_attn(const float* __restrict__ ft,
                         const float* __restrict__ al, const float* __restrict__ alb,
                         const float* __restrict__ ar, const float* __restrict__ arb,
                         float* __restrict__ a1, float* __restrict__ a2,
                         int N, int H, int hid)
{
    int t = blockIdx.x * blockDim.x + threadIdx.x;
    if (t >= N * H) return;
    int n = t / H, h = t % H;
    const float* f  = ft + (size_t)n * H * hid + h * hid;
    const float* lp = al + h * hid;
    const float* rp = ar + h * hid;
    float s1 = 0.f, s2 = 0.f;
    for (int d = 0; d < hid; d += 4) {
        float4 fv = *(const float4*)(f + d);
        float4 lv = *(const float4*)(lp + d);
        float4 rv = *(const float4*)(rp + d);
        s1 += fv.x*lv.x + fv.y*lv.y + fv.z*lv.z + fv.w*lv.w;
        s2 += fv.x*rv.x + fv.y*rv.y + fv.z*rv.z + fv.w*rv.w;
    }
    a1[t] = s1 + alb[h];
    a2[t] = s2 + arb[h];
}

// w[e,h] = exp(leaky_relu(a2[src]+a1[dst])); denom[dst,h] += w
__global__ void gat_edge_softmax(const int* __restrict__ src, const int* __restrict__ dst,
                                 const float* __restrict__ a1, const float* __restrict__ a2,
                                 float* __restrict__ w, float* __restrict__ denom,
                                 int E, int H)
{
    int t = blockIdx.x * blockDim.x + threadIdx.x;
    if (t >= E * H) return;
    int e = t / H, h = t % H;
    int s = src[e], d = dst[e];
    float v = a2[(size_t)s * H + h] + a1[(size_t)d * H + h];
    v = (v > 0.f) ? v : 0.01f * v;           // leaky_relu slope 0.01
    float wv = expf(v);
    w[t] = wv;
    atomicAdd(&denom[(size_t)d * H + h], wv);
}

// agg[dst, c] += ft[src, c] * (w[e, c/hid] / denom[dst, c/hid]); one block per edge
__global__ void gat_edge_aggregate(const int* __restrict__ src, const int* __restrict__ dst,
                                   const float* __restrict__ w, const float* __restrict__ denom,
                                   const float* __restrict__ ft, float* __restrict__ agg,
                                   int C, int hid, int H)
{
    int e = blockIdx.x;
    int c = threadIdx.x;
    if (c >= C) return;
    int h = c / hid;
    int s = src[e], d = dst[e];          // uniform -> scalar loads
    float att = w[(size_t)e * H + h] / denom[(size_t)d * H + h];
    atomicAdd(&agg[(size_t)d * C + c], ft[(size_t)s * C + c] * att);
}

__global__ void gat_zero(float* __restrict__ p, size_t n)
{
    size_t i = (size_t)blockIdx.x * blockDim.x + threadIdx.x;
    if (i < n) p[i] = 0.f;
}

// x[i] = elu(x[i] + (r ? r[i] : 0))
__global__ void gat_elu_add(float* __restrict__ x, const float* __restrict__ r, size_t n)
{
    size_t i = (size_t)blockIdx.x * blockDim.x + threadIdx.x;
    if (i >= n) return;
    float v = x[i] + (r ? r[i] : 0.f);
    x[i] = (v > 0.f) ? v : expm1f(v);
}

extern "C" void kernel_launch(void* const* d_in, const int* in_sizes, int n_in,
                              void* d_out, int out_size, void* d_ws, size_t ws_size,
                              hipStream_t stream)
{
    const float* features = (const float*)d_in[0];
    const int*   src      = (const int*)d_in[1];
    const int*   dst      = (const int*)d_in[2];
    const float* W0  = (const float*)d_in[3];
    const float* b0  = (const float*)d_in[4];
    const float* al0 = (const float*)d_in[5];
    const float* alb0= (const float*)d_in[6];
    const float* ar0 = (const float*)d_in[7];
    const float* arb0= (const float*)d_in[8];
    const float* W1  = (const float*)d_in[9];
    const float* b1  = (const float*)d_in[10];
    const float* al1 = (const float*)d_in[11];
    const float* alb1= (const float*)d_in[12];
    const float* ar1 = (const float*)d_in[13];
    const float* arb1= (const float*)d_in[14];
    const float* R1  = (const float*)d_in[15];
    const float* Rb1 = (const float*)d_in[16];
    const float* Wf  = (const float*)d_in[17];
    const float* bf  = (const float*)d_in[18];
    const float* alf = (const float*)d_in[19];
    const float* alfb= (const float*)d_in[20];
    const float* arf = (const float*)d_in[21];
    const float* arfb= (const float*)d_in[22];
    const float* Rf  = (const float*)d_in[23];
    const float* Rbf = (const float*)d_in[24];

    const int N = in_sizes[0] / DIN;     // 50000
    const int E = in_sizes[1];           // 800000
    const int C = HEADS * DHID;          // 256
    const int PWF = 64;                  // padded panel width for CLS=40

    float* ws = (float*)d_ws;
    size_t off = 0;
    float* bp0   = ws + off; off += (size_t)DIN * C;    // packed W0   [K x 256]
    float* bp1   = ws + off; off += (size_t)C * C;      // packed W1
    float* bpR   = ws + off; off += (size_t)C * C;      // packed R1
    float* bpF   = ws + off; off += (size_t)C * PWF;    // packed Wf   [K x 64]
    float* bpRf  = ws + off; off += (size_t)C * PWF;    // packed Rf
    float* a1    = ws + off; off += (size_t)N * HEADS;
    float* a2    = ws + off; off += (size_t)N * HEADS;
    float* den   = ws + off; off += (size_t)N * HEADS;
    float* wbuf  = ws + off; off += (size_t)E * HEADS;
    float* ft    = ws + off; off += (size_t)N * C;      // reused every layer
    float* xb    = ws + off; off += (size_t)N * C;      // x / agg buffer
    float* resid = ws + off; off += (size_t)N * C;      // resid (layer1 & final)
    float* outp  = (float*)d_out;

    const dim3 B256(256);
    const int gemmGrid = (N + 15) / 16;

    // pack weight panels (pair-major, zero-padded)
    gat_pack<<<dim3((DIN * C + 255) / 256), B256, 0, stream>>>(W0, bp0, DIN, DHID, HEADS, C);
    gat_pack<<<dim3((C * C + 255) / 256),   B256, 0, stream>>>(W1, bp1, C, DHID, HEADS, C);
    gat_pack<<<dim3((C * C + 255) / 256),   B256, 0, stream>>>(R1, bpR, C, DHID, HEADS, C);
    gat_pack<<<dim3((C * PWF + 255) / 256), B256, 0, stream>>>(Wf, bpF, C, DCLS, 1, PWF);
    gat_pack<<<dim3((C * PWF + 255) / 256), B256, 0, stream>>>(Rf, bpRf, C, DCLS, 1, PWF);

    // ---------------- layer 0 (no residual) ----------------
    gat_gemm_wmma<<<dim3(gemmGrid), dim3(128), 0, stream>>>(features, bp0, b0, ft, N, DIN, C, C);
    gat_attn<<<dim3((N * HEADS + 255) / 256), B256, 0, stream>>>(ft, al0, alb0, ar0, arb0, a1, a2, N, HEADS, DHID);
    gat_zero<<<dim3((N * HEADS + 255) / 256), B256, 0, stream>>>(den, (size_t)N * HEADS);
    gat_edge_softmax<<<dim3((E * HEADS + 255) / 256), B256, 0, stream>>>(src, dst, a1, a2, wbuf, den, E, HEADS);
    gat_zero<<<dim3(((size_t)N * C + 255) / 256), B256, 0, stream>>>(xb, (size_t)N * C);
    gat_edge_aggregate<<<dim3(E), dim3(C), 0, stream>>>(src, dst, wbuf, den, ft, xb, C, DHID, HEADS);
    gat_elu_add<<<dim3(((size_t)N * C + 255) / 256), B256, 0, stream>>>(xb, (const float*)nullptr, (size_t)N * C);

    // ---------------- layer 1 (residual fc) ----------------
    gat_gemm_wmma<<<dim3(gemmGrid), dim3(128), 0, stream>>>(xb, bp1, b1,  ft,    N, C, C, C);
    gat_gemm_wmma<<<dim3(gemmGrid), dim3(128), 0, stream>>>(xb, bpR, Rb1, resid, N, C, C, C);
    gat_attn<<<dim3((N * HEADS + 255) / 256), B256, 0, stream>>>(ft, al1, alb1, ar1, arb1, a1, a2, N, HEADS, DHID);
    gat_zero<<<dim3((N * HEADS + 255) / 256), B256, 0, stream>>>(den, (size_t)N * HEADS);
    gat_edge_softmax<<<dim3((E * HEADS + 255) / 256), B256, 0, stream>>>(src, dst, a1, a2, wbuf, den, E, HEADS);
    gat_zero<<<dim3(((size_t)N * C + 255) / 256), B256, 0, stream>>>(xb, (size_t)N * C);   // x1 dead after GEMMs
    gat_edge_aggregate<<<dim3(E), dim3(C), 0, stream>>>(src, dst, wbuf, den, ft, xb, C, DHID, HEADS);
    gat_elu_add<<<dim3(((size_t)N * C + 255) / 256), B256, 0, stream>>>(xb, resid, (size_t)N * C);

    // ---------------- final layer (single head -> CLS, residual fc) --------
    gat_gemm_wmma<<<dim3(gemmGrid), dim3(128), 0, stream>>>(xb, bpF,  bf,  ft,    N, C, PWF, DCLS);
    gat_gemm_wmma<<<dim3(gemmGrid), dim3(128), 0, stream>>>(xb, bpRf, Rbf, resid, N, C, PWF, DCLS);
    gat_attn<<<dim3((N + 255) / 256), B256, 0, stream>>>(ft, alf, alfb, arf, arfb, a1, a2, N, 1, DCLS);
    gat_zero<<<dim3((N + 255) / 256), B256, 0, stream>>>(den, (size_t)N);
    gat_edge_softmax<<<dim3((E + 255) / 256), B256, 0, stream>>>(src, dst, a1, a2, wbuf, den, E, 1);
    gat_zero<<<dim3(((size_t)N * DCLS + 255) / 256), B256, 0, stream>>>(outp, (size_t)N * DCLS);
    gat_edge_aggregate<<<dim3(E), dim3(64), 0, stream>>>(src, dst, wbuf, den, ft, outp, DCLS, DCLS, 1);
    gat_elu_add<<<dim3(((size_t)N * DCLS + 255) / 256), B256, 0, stream>>>(outp, resid, (size_t)N * DCLS);

    (void)n_in; (void)out_size; (void)ws_size;
}